// Model_83614423318751
// MI455X (gfx1250) — compile-verified
//
#include <hip/hip_runtime.h>

#define NN   20000
#define EE   640000
#define ENE  660000   // E + N (self loops)
#define INF_ 2000
#define CC   64
#define HH   4
#define HC   256

typedef __attribute__((ext_vector_type(2))) float v2f;
typedef __attribute__((ext_vector_type(8))) float v8f;
typedef int v4i_gcc __attribute__((__vector_size__(4 * sizeof(int))));

__device__ __forceinline__ v8f wmma_f32(v2f a, v2f b, v8f c) {
  return __builtin_amdgcn_wmma_f32_16x16x4_f32(false, a, false, b, (short)0, c,
                                               false, false);
}

// ---------------------------------------------------------------------------
// Async global->LDS staging (CDNA5). Falls back to direct LDS stores if the
// builtin is unavailable (guarded so compilation can never break).
// Param types per hipcc diagnostic: (v4i AS1*, v4i AS3*, imm, imm).
// ---------------------------------------------------------------------------
#if defined(__has_builtin)
#if __has_builtin(__builtin_amdgcn_global_load_async_to_lds_b128)
#define HAVE_ASYNC 1
#else
#define HAVE_ASYNC 0
#endif
#else
#define HAVE_ASYNC 0
#endif

__device__ __forceinline__ void cp16(const float* g, float* l) {
#if HAVE_ASYNC
  __builtin_amdgcn_global_load_async_to_lds_b128(
      (__attribute__((address_space(1))) v4i_gcc*)(unsigned long long)(
          uintptr_t)g,
      (__attribute__((address_space(3))) v4i_gcc*)(unsigned)(uintptr_t)l, 0, 0);
#else
  *(float4*)l = *(const float4*)g;
#endif
}

__device__ __forceinline__ void wait_async0() {
#if HAVE_ASYNC
#if __has_builtin(__builtin_amdgcn_s_wait_asynccnt)
  __builtin_amdgcn_s_wait_asynccnt(0);
#else
  asm volatile("s_wait_asynccnt 0x0" ::: "memory");
#endif
#endif
}
__device__ __forceinline__ void wait_async3() {
#if HAVE_ASYNC
#if __has_builtin(__builtin_amdgcn_s_wait_asynccnt)
  __builtin_amdgcn_s_wait_asynccnt(3);
#else
  asm volatile("s_wait_asynccnt 0x3" ::: "memory");
#endif
#endif
}

// ---------------------------------------------------------------------------
// K1: split-K encoder GEMM  h_part[s] = x[:, s*400:(s+1)*400] @ W_ae[slice]
// Block: 8 waves -> 128 rows x 64 cols. Grid (157, 5). Double-buffered async
// LDS staging, chunk = 16 K-values (3 x b128 per thread per chunk, uniform).
// ---------------------------------------------------------------------------
#define SPLITK 5
#define KR 400
#define EC 16
#define EC_NC 25      // chunks per split
#define EC_XS 20      // x tile LDS stride (floats), rows 16B aligned
#define EC_WS 68      // W tile LDS stride (floats), rows 16B aligned

__device__ __forceinline__ void k1_issue(const float* __restrict__ x,
                                         const float* __restrict__ W,
                                         int rowBase, int k0, int tid,
                                         float* sx, float* sw) {
#pragma unroll
  for (int q = 0; q < 3; ++q) {
    int i = tid + 256 * q;
    if (i < 512) {                       // x tile: 128 rows x 16 floats
      int r = i >> 2, c4 = (i & 3) * 4;
      int gr = rowBase + r;
      gr = gr < NN ? gr : NN - 1;        // clamp: garbage rows never stored
      cp16(x + (long)gr * INF_ + k0 + c4, sx + r * EC_XS + c4);
    } else {                             // W tile: 16 rows x 64 floats
      int iw = i - 512;
      int r = iw >> 4, c4 = (iw & 15) * 4;
      cp16(W + (k0 + r) * CC + c4, sw + r * EC_WS + c4);
    }
  }
}

__global__ __launch_bounds__(256) void k_encoder(const float* __restrict__ x,
                                                 const float* __restrict__ W,
                                                 float* __restrict__ hparts) {
  __shared__ float sx[2][128 * EC_XS];   // 2 x 10240 B
  __shared__ float sw[2][EC * EC_WS];    // 2 x 4352 B
  const int tid = threadIdx.x;
  const int lane = tid & 31, wave = tid >> 5;
  const int half = lane >> 4, n = lane & 15;
  const int rowBase = blockIdx.x * 128;
  const int waveRow = wave * 16;
  const int kbase = blockIdx.y * KR;

  v8f acc[4];
#pragma unroll
  for (int t = 0; t < 4; ++t)
#pragma unroll
    for (int r = 0; r < 8; ++r) acc[t][r] = 0.f;

  k1_issue(x, W, rowBase, kbase, tid, sx[0], sw[0]);
  for (int ci = 0; ci < EC_NC; ++ci) {
    const int cur = ci & 1;
    if (ci + 1 < EC_NC) {
      k1_issue(x, W, rowBase, kbase + (ci + 1) * EC, tid, sx[cur ^ 1],
               sw[cur ^ 1]);
      wait_async3();                     // my chunk-ci copies have landed
    } else {
      wait_async0();
    }
    __syncthreads();                     // everyone's chunk-ci copies landed
    const float* bx = sx[cur];
    const float* bw = sw[cur];
#pragma unroll
    for (int kk = 0; kk < EC; kk += 4) {
      int kA = kk + 2 * half;
      v2f a;
      a.x = bx[(waveRow + n) * EC_XS + kA];
      a.y = bx[(waveRow + n) * EC_XS + kA + 1];
#pragma unroll
      for (int t = 0; t < 4; ++t) {
        v2f bf;
        bf.x = bw[kA * EC_WS + t * 16 + n];
        bf.y = bw[(kA + 1) * EC_WS + t * 16 + n];
        acc[t] = wmma_f32(a, bf, acc[t]);
      }
    }
    __syncthreads();                     // safe to overwrite this buffer
  }

  float* hp = hparts + (size_t)blockIdx.y * NN * CC;
#pragma unroll
  for (int t = 0; t < 4; ++t)
#pragma unroll
    for (int r = 0; r < 8; ++r) {
      int row = rowBase + waveRow + r + 8 * half;
      if (row < NN) hp[row * CC + t * 16 + n] = acc[t][r];
    }
}

// h = sum of split-K partials + bias (deterministic combine)
__global__ __launch_bounds__(256) void k_combine(const float* __restrict__ hp,
                                                 const float* __restrict__ bias,
                                                 float* __restrict__ h) {
  int t = blockIdx.x * 256 + threadIdx.x;
  if (t >= NN * CC) return;
  float s = bias[t & 63];
#pragma unroll
  for (int k = 0; k < SPLITK; ++k) s += hp[(size_t)k * NN * CC + t];
  h[t] = s;
}

// ---------------------------------------------------------------------------
// K2: y = h @ W + b   [20000 x 64] * [64 x 256]   (xl and xr)
// ---------------------------------------------------------------------------
#define P_ROWS 32
#define P_HS 66
#define P_WS 258

__global__ __launch_bounds__(256) void k_proj(const float* __restrict__ hin,
                                              const float* __restrict__ W,
                                              const float* __restrict__ bias,
                                              float* __restrict__ out) {
  __shared__ float sh[P_ROWS * P_HS];
  __shared__ float sw[32 * P_WS];
  const int tid = threadIdx.x;
  const int lane = tid & 31, wave = tid >> 5;
  const int half = lane >> 4, n = lane & 15;
  const int rowBase = blockIdx.x * P_ROWS;
  const int waveRow = (wave >> 2) * 16;
  const int waveCol = (wave & 3) * 64;

  for (int i = tid; i < P_ROWS * CC; i += 256) {
    int r = i >> 6, c = i & 63;
    int gr = rowBase + r;
    sh[r * P_HS + c] = (gr < NN) ? hin[gr * CC + c] : 0.f;
  }

  v8f acc[4];
#pragma unroll
  for (int t = 0; t < 4; ++t)
#pragma unroll
    for (int r = 0; r < 8; ++r) acc[t][r] = 0.f;

  for (int kc = 0; kc < CC; kc += 32) {
    __syncthreads();
    for (int i = tid; i < 32 * HC; i += 256) {
      int r = i >> 8, c = i & 255;
      sw[r * P_WS + c] = W[(kc + r) * HC + c];
    }
    __syncthreads();
#pragma unroll
    for (int kk = 0; kk < 32; kk += 4) {
      int kA = kk + 2 * half;
      v2f a;
      a.x = sh[(waveRow + n) * P_HS + kc + kA];
      a.y = sh[(waveRow + n) * P_HS + kc + kA + 1];
#pragma unroll
      for (int t = 0; t < 4; ++t) {
        v2f bf;
        bf.x = sw[kA * P_WS + waveCol + t * 16 + n];
        bf.y = sw[(kA + 1) * P_WS + waveCol + t * 16 + n];
        acc[t] = wmma_f32(a, bf, acc[t]);
      }
    }
  }
#pragma unroll
  for (int t = 0; t < 4; ++t)
#pragma unroll
    for (int r = 0; r < 8; ++r) {
      int row = rowBase + waveRow + r + 8 * half;
      int col = waveCol + t * 16 + n;
      if (row < NN) out[(long)row * HC + col] = acc[t][r] + bias[col];
    }
}

// ---------------------------------------------------------------------------
// CSR build: degree histogram -> exclusive scan -> fill src lists
// ---------------------------------------------------------------------------
__global__ __launch_bounds__(256) void k_count(const int* __restrict__ ei,
                                               int* __restrict__ deg) {
  int e = blockIdx.x * 256 + threadIdx.x;
  if (e >= ENE) return;
  int dst = (e < EE) ? ei[EE + e] : e - EE;
  atomicAdd(&deg[dst], 1);
}

__global__ __launch_bounds__(1024) void k_scan(const int* __restrict__ deg,
                                               int* __restrict__ rowptr) {
  __shared__ int sbuf[1024];
  __shared__ int scarry;
  const int tid = threadIdx.x;
  if (tid == 0) scarry = 0;
  __syncthreads();
  for (int base = 0; base < NN; base += 1024) {
    int v = (base + tid < NN) ? deg[base + tid] : 0;
    int a = v;
    sbuf[tid] = a;
    __syncthreads();
    for (int off = 1; off < 1024; off <<= 1) {
      int t = (tid >= off) ? sbuf[tid - off] : 0;
      __syncthreads();
      a += t;
      sbuf[tid] = a;
      __syncthreads();
    }
    int carry = scarry;
    if (base + tid < NN) rowptr[base + tid] = carry + a - v;  // exclusive
    __syncthreads();
    if (tid == 1023) scarry = carry + sbuf[1023];
    __syncthreads();
  }
  if (tid == 0) rowptr[NN] = scarry;
}

__global__ __launch_bounds__(256) void k_fill(const int* __restrict__ ei,
                                              const int* __restrict__ rowptr,
                                              int* __restrict__ cursor,
                                              int* __restrict__ esrc) {
  int e = blockIdx.x * 256 + threadIdx.x;
  if (e >= ENE) return;
  int src, dst;
  if (e < EE) { src = ei[e]; dst = ei[EE + e]; }
  else { src = e - EE; dst = src; }
  int pos = rowptr[dst] + atomicAdd(&cursor[dst], 1);
  esrc[pos] = src;
}

// ---------------------------------------------------------------------------
// Fused GATv2 attention, dst-centric, online softmax. One wave per node.
// Lane l -> head l>>3, 8 channels (l&7)*8.. ; single gather pass over xl[src];
// no float atomics anywhere.
// ---------------------------------------------------------------------------
__global__ __launch_bounds__(256) void k_attn(const int* __restrict__ rowptr,
                                              const int* __restrict__ esrc,
                                              const float* __restrict__ xl,
                                              const float* __restrict__ xr,
                                              const float* __restrict__ att,
                                              float* __restrict__ agg) {
  int node = (int)((blockIdx.x * 256u + threadIdx.x) >> 5);
  if (node >= NN) return;
  const int lane = threadIdx.x & 31;
  const int off = (lane >> 3) * CC + (lane & 7) * 8;

  float4 r0 = *(const float4*)(xr + (long)node * HC + off);
  float4 r1 = *(const float4*)(xr + (long)node * HC + off + 4);
  float4 w0 = *(const float4*)(att + off);
  float4 w1 = *(const float4*)(att + off + 4);

  float m = -3.0e38f, denom = 0.f;
  float4 acc0 = make_float4(0.f, 0.f, 0.f, 0.f);
  float4 acc1 = make_float4(0.f, 0.f, 0.f, 0.f);

  int p0 = __builtin_amdgcn_readfirstlane(rowptr[node]);
  int p1 = __builtin_amdgcn_readfirstlane(rowptr[node + 1]);
  for (int p = p0; p < p1; ++p) {
    int j = __builtin_amdgcn_readfirstlane(esrc[p]);
    float4 a0 = *(const float4*)(xl + (long)j * HC + off);
    float4 a1 = *(const float4*)(xl + (long)j * HC + off + 4);
    float g, s = 0.f;
    g = a0.x + r0.x; g = g > 0.f ? g : 0.2f * g; s += g * w0.x;
    g = a0.y + r0.y; g = g > 0.f ? g : 0.2f * g; s += g * w0.y;
    g = a0.z + r0.z; g = g > 0.f ? g : 0.2f * g; s += g * w0.z;
    g = a0.w + r0.w; g = g > 0.f ? g : 0.2f * g; s += g * w0.w;
    g = a1.x + r1.x; g = g > 0.f ? g : 0.2f * g; s += g * w1.x;
    g = a1.y + r1.y; g = g > 0.f ? g : 0.2f * g; s += g * w1.y;
    g = a1.z + r1.z; g = g > 0.f ? g : 0.2f * g; s += g * w1.z;
    g = a1.w + r1.w; g = g > 0.f ? g : 0.2f * g; s += g * w1.w;
    s += __shfl_xor(s, 1, 32);
    s += __shfl_xor(s, 2, 32);
    s += __shfl_xor(s, 4, 32);          // all 8 lanes of head group hold logit
    float mn = fmaxf(m, s);
    float scale = __expf(m - mn);       // first iter: exp(-huge) == 0
    float e = __expf(s - mn);
    denom = denom * scale + e;
    acc0.x = acc0.x * scale + e * a0.x;
    acc0.y = acc0.y * scale + e * a0.y;
    acc0.z = acc0.z * scale + e * a0.z;
    acc0.w = acc0.w * scale + e * a0.w;
    acc1.x = acc1.x * scale + e * a1.x;
    acc1.y = acc1.y * scale + e * a1.y;
    acc1.z = acc1.z * scale + e * a1.z;
    acc1.w = acc1.w * scale + e * a1.w;
    m = mn;
  }
  float inv = 1.f / denom;
  float* po = agg + (long)node * HC + off;
  *(float4*)po = make_float4(acc0.x * inv, acc0.y * inv, acc0.z * inv,
                             acc0.w * inv);
  *(float4*)(po + 4) = make_float4(acc1.x * inv, acc1.y * inv, acc1.z * inv,
                                   acc1.w * inv);
}

// ---------------------------------------------------------------------------
// K6: a = relu((agg + b_conv) @ W_lin + b_lin); y = a + h; LayerNorm -> out
// ---------------------------------------------------------------------------
#define F_ROWS 128
#define F_CHUNK 64
#define F_S 66

__global__ __launch_bounds__(256) void k_final(
    const float* __restrict__ agg, const float* __restrict__ bconv,
    const float* __restrict__ Wlin, const float* __restrict__ blin,
    const float* __restrict__ hres, const float* __restrict__ lnw,
    const float* __restrict__ lnb, float* __restrict__ outp) {
  __shared__ float sa[F_ROWS * F_S];
  __shared__ float sw[F_CHUNK * F_S];
  const int tid = threadIdx.x;
  const int lane = tid & 31, wave = tid >> 5;
  const int half = lane >> 4, n = lane & 15;
  const int rowBase = blockIdx.x * F_ROWS;
  const int waveRow = wave * 16;

  v8f acc[4];
#pragma unroll
  for (int t = 0; t < 4; ++t)
#pragma unroll
    for (int r = 0; r < 8; ++r) acc[t][r] = 0.f;

  for (int k0 = 0; k0 < HC; k0 += F_CHUNK) {
    __syncthreads();
    for (int i = tid; i < F_ROWS * F_CHUNK; i += 256) {
      int r = i >> 6, c = i & 63;
      int gr = rowBase + r;
      sa[r * F_S + c] =
          (gr < NN) ? agg[(long)gr * HC + k0 + c] + bconv[k0 + c] : 0.f;
    }
    for (int i = tid; i < F_CHUNK * CC; i += 256) {
      int r = i >> 6, c = i & 63;
      sw[r * F_S + c] = Wlin[(k0 + r) * CC + c];
    }
    __syncthreads();
#pragma unroll 4
    for (int kk = 0; kk < F_CHUNK; kk += 4) {
      int kA = kk + 2 * half;
      v2f a;
      a.x = sa[(waveRow + n) * F_S + kA];
      a.y = sa[(waveRow + n) * F_S + kA + 1];
#pragma unroll
      for (int t = 0; t < 4; ++t) {
        v2f bf;
        bf.x = sw[kA * F_S + t * 16 + n];
        bf.y = sw[(kA + 1) * F_S + t * 16 + n];
        acc[t] = wmma_f32(a, bf, acc[t]);
      }
    }
  }

  float bl4[4], lw4[4], lb4[4];
#pragma unroll
  for (int t = 0; t < 4; ++t) {
    int col = t * 16 + n;
    bl4[t] = blin[col]; lw4[t] = lnw[col]; lb4[t] = lnb[col];
  }
#pragma unroll
  for (int r = 0; r < 8; ++r) {
    int row = rowBase + waveRow + r + 8 * half;
    bool ok = row < NN;
    float y[4]; float s = 0.f;
#pragma unroll
    for (int t = 0; t < 4; ++t) {
      float a = acc[t][r] + bl4[t];
      a = a > 0.f ? a : 0.f;
      float hv = ok ? hres[row * CC + t * 16 + n] : 0.f;
      y[t] = a + hv;
      s += y[t];
    }
    s += __shfl_xor(s, 1, 32); s += __shfl_xor(s, 2, 32);
    s += __shfl_xor(s, 4, 32); s += __shfl_xor(s, 8, 32);
    float mean = s * 0.015625f;
    float vs = 0.f;
#pragma unroll
    for (int t = 0; t < 4; ++t) { float d = y[t] - mean; vs += d * d; }
    vs += __shfl_xor(vs, 1, 32); vs += __shfl_xor(vs, 2, 32);
    vs += __shfl_xor(vs, 4, 32); vs += __shfl_xor(vs, 8, 32);
    float inv = rsqrtf(vs * 0.015625f + 1e-12f);
#pragma unroll
    for (int t = 0; t < 4; ++t)
      if (ok)
        outp[row * CC + t * 16 + n] = lw4[t] * (y[t] - mean) * inv + lb4[t];
  }
}

// ---------------------------------------------------------------------------
extern "C" void kernel_launch(void* const* d_in, const int* in_sizes, int n_in,
                              void* d_out, int out_size, void* d_ws,
                              size_t ws_size, hipStream_t stream) {
  const float* x     = (const float*)d_in[0];
  const int*   ei    = (const int*)d_in[1];
  const float* W_ae  = (const float*)d_in[2];
  const float* b_ae  = (const float*)d_in[3];
  const float* Wl    = (const float*)d_in[4];
  const float* bl    = (const float*)d_in[5];
  const float* Wr    = (const float*)d_in[6];
  const float* br    = (const float*)d_in[7];
  const float* att   = (const float*)d_in[8];
  const float* bconv = (const float*)d_in[9];
  const float* Wlin  = (const float*)d_in[10];
  const float* blin  = (const float*)d_in[11];
  const float* lnw   = (const float*)d_in[12];
  const float* lnb   = (const float*)d_in[13];
  float* out = (float*)d_out;

  float* ws = (float*)d_ws;
  float* h   = ws;                          // N*C
  float* xl  = h + (size_t)NN * CC;         // N*HC
  float* xr  = xl + (size_t)NN * HC;        // N*HC
  float* agg = xr + (size_t)NN * HC;        // N*HC
  // split-K partials (SPLITK * N*C = 6.4M floats) overlap xl+xr (10.24M):
  // consumed by k_combine before k_proj writes xl/xr.
  float* hparts = xl;
  int* ibuf   = (int*)(agg + (size_t)NN * HC);
  int* deg    = ibuf;                       // NN
  int* rowptr = ibuf + NN;                  // NN+1
  int* cursor = ibuf + 2 * NN + 1;          // NN
  int* esrc   = ibuf + 3 * NN + 1;          // ENE

  (void)hipMemsetAsync(deg, 0, NN * sizeof(int), stream);
  (void)hipMemsetAsync(cursor, 0, NN * sizeof(int), stream);

  // CSR build (only needs edge_index)
  k_count<<<(ENE + 255) / 256, 256, 0, stream>>>(ei, deg);
  k_scan<<<1, 1024, 0, stream>>>(deg, rowptr);
  k_fill<<<(ENE + 255) / 256, 256, 0, stream>>>(ei, rowptr, cursor, esrc);

  // Encoder GEMM (split-K, async double-buffered LDS staging)
  k_encoder<<<dim3((NN + 127) / 128, SPLITK), 256, 0, stream>>>(x, W_ae,
                                                                hparts);
  k_combine<<<(NN * CC + 255) / 256, 256, 0, stream>>>(hparts, b_ae, h);

  // Projections (overwrite hparts region only after combine)
  k_proj<<<(NN + P_ROWS - 1) / P_ROWS, 256, 0, stream>>>(h, Wl, bl, xl);
  k_proj<<<(NN + P_ROWS - 1) / P_ROWS, 256, 0, stream>>>(h, Wr, br, xr);

  // Fused attention (dst-centric, online softmax, no float atomics)
  k_attn<<<(NN + 7) / 8, 256, 0, stream>>>(rowptr, esrc, xl, xr, att, agg);

  // Output GEMM + relu + residual + LayerNorm
  k_final<<<(NN + F_ROWS - 1) / F_ROWS, 256, 0, stream>>>(
      agg, bconv, Wlin, blin, h, lnw, lnb, out);
}